// PointsetGrouper_55954833933050
// MI455X (gfx1250) — compile-verified
//
#include <hip/hip_runtime.h>

// ---------------------------------------------------------------------------
// PointsetGrouper for MI455X (gfx1250, wave32)
//   B=4, N=8192, C=128, S=2048, K=32, radius=0.2
// Pipeline (all on `stream`, stream order gives dependencies):
//   1) fps_kernel    : 1 workgroup per batch, persistent, register-resident
//                      dists, wave32 shuffle argmax. Writes fps_idx + new_xyz.
//   2) ballq_kernel  : pairwise distances via V_WMMA_F32_16X16X4_F32
//                      (16 queries x 16 points per wave, K-dim = 3 coords),
//                      in-radius mask -> LDS, ordered first-K index scan with
//                      uniform early exit. Writes bq_idx.
//   3) group_kernel  : gather K rows of C channels, affine + max over K,
//                      transposed store. HBM-bound (~134 MB -> ~6us @23.3TB/s).
// ---------------------------------------------------------------------------

typedef __attribute__((ext_vector_type(2))) float v2f;
typedef __attribute__((ext_vector_type(8))) float v8f;

namespace {
constexpr int   B_  = 4;
constexpr int   N_  = 8192;
constexpr int   C_  = 128;
constexpr int   K_  = 32;
constexpr int   S_  = 2048;               // N / REDUCE
constexpr float R2_ = 0.2f * 0.2f;

constexpr int FPS_THREADS = 1024;
constexpr int PPT         = N_ / FPS_THREADS;  // 8 points per thread
constexpr int QT          = 16;                // queries per ball-query block
constexpr int CHUNK       = 128;               // 8 waves * 16 columns
}

// ---------------------------------------------------------------------------
// Kernel 1: furthest point sampling (one workgroup per batch)
// ---------------------------------------------------------------------------
__global__ __launch_bounds__(FPS_THREADS)
void fps_kernel(const float* __restrict__ xyz,
                int* __restrict__ fps_idx,
                float* __restrict__ new_xyz)
{
    const int b   = blockIdx.x;
    const int tid = threadIdx.x;
    const float* bx = xyz + (size_t)b * N_ * 3;

    // register-resident coords and running min-distance for owned points
    float px[PPT], py[PPT], pz[PPT], dist[PPT];
#pragma unroll
    for (int i = 0; i < PPT; ++i) {
        const int p = tid + i * FPS_THREADS;       // strided ownership
        px[i]  = bx[3 * p + 0];
        py[i]  = bx[3 * p + 1];
        pz[i]  = bx[3 * p + 2];
        dist[i] = 1e10f;
    }

    __shared__ float s_cent[3];
    __shared__ float s_rv[32];
    __shared__ int   s_ri[32];
    __shared__ int   s_far;

    const int lane = tid & 31;
    const int wave = tid >> 5;

    int farthest = 0;
    for (int s = 0; s < S_; ++s) {
        // owner of the current farthest point broadcasts its coords
        if (tid == (farthest & (FPS_THREADS - 1))) {
            const int slot = farthest / FPS_THREADS;
            s_cent[0] = px[slot]; s_cent[1] = py[slot]; s_cent[2] = pz[slot];
        }
        __syncthreads();
        const float cx = s_cent[0], cy = s_cent[1], cz = s_cent[2];
        if (tid == 0) {
            fps_idx[b * S_ + s] = farthest;
            float* o = new_xyz + ((size_t)b * S_ + s) * 3;
            o[0] = cx; o[1] = cy; o[2] = cz;
        }

        // update min-distances, track local argmax (lowest index on ties)
        float bv = -1.0f; int bi = 0;
#pragma unroll
        for (int i = 0; i < PPT; ++i) {
            const float dx = px[i] - cx, dy = py[i] - cy, dz = pz[i] - cz;
            const float d = dx * dx + dy * dy + dz * dz;
            dist[i] = fminf(dist[i], d);
            const int p = tid + i * FPS_THREADS;
            if (dist[i] > bv || (dist[i] == bv && p < bi)) { bv = dist[i]; bi = p; }
        }
        // wave32 shuffle argmax
#pragma unroll
        for (int off = 16; off > 0; off >>= 1) {
            const float ov = __shfl_down(bv, off, 32);
            const int   oi = __shfl_down(bi, off, 32);
            if (ov > bv || (ov == bv && oi < bi)) { bv = ov; bi = oi; }
        }
        if (lane == 0) { s_rv[wave] = bv; s_ri[wave] = bi; }
        __syncthreads();
        if (wave == 0) {
            bv = s_rv[lane]; bi = s_ri[lane];
#pragma unroll
            for (int off = 16; off > 0; off >>= 1) {
                const float ov = __shfl_down(bv, off, 32);
                const int   oi = __shfl_down(bi, off, 32);
                if (ov > bv || (ov == bv && oi < bi)) { bv = ov; bi = oi; }
            }
            if (lane == 0) s_far = bi;
        }
        __syncthreads();
        farthest = s_far;
    }
}

// ---------------------------------------------------------------------------
// Kernel 2: ball query via f32 WMMA distance tiles
//   block = 256 threads (8 waves), handles 16 queries; each wave computes a
//   16x16 Q.X^T tile with V_WMMA_F32_16X16X4_F32 (K-dim = xyz padded to 4).
// ---------------------------------------------------------------------------
__global__ __launch_bounds__(256)
void ballq_kernel(const float* __restrict__ xyz,
                  const float* __restrict__ new_xyz,
                  int* __restrict__ bq_idx)
{
    const int blk   = blockIdx.x;
    const int b     = blk / (S_ / QT);
    const int qbase = (blk % (S_ / QT)) * QT;
    const int tid   = threadIdx.x;
    const int lane  = tid & 31;
    const int w     = tid >> 5;
    const int m0    = lane & 15;
    const int half8 = (lane >> 4) << 3;        // 0 or 8: row offset of D half

    __shared__ float qx[QT], qy[QT], qz[QT], q2[QT];
    __shared__ unsigned char s_mask[QT * CHUNK];
    __shared__ int s_cnt[QT];
    __shared__ int s_done;

    if (tid < QT) {
        const float* qp = new_xyz + ((size_t)(b * S_ + qbase + tid)) * 3;
        const float x = qp[0], y = qp[1], z = qp[2];
        qx[tid] = x; qy[tid] = y; qz[tid] = z;
        q2[tid] = x * x + y * y + z * z;
        s_cnt[tid] = 0;
    }
    __syncthreads();

    // A operand (queries) is loop-invariant:
    //   lanes 0-15 : K=0,1 -> (qx,qy); lanes 16-31 : K=2,3 -> (qz,0)
    v2f Av;
    Av.x = (lane < 16) ? qx[m0] : qz[m0];
    Av.y = (lane < 16) ? qy[m0] : 0.0f;
    float q2r[8];
#pragma unroll
    for (int j = 0; j < 8; ++j) q2r[j] = q2[j + half8];

    int cnt = 0;
    const int base = (b * S_ + qbase + tid) * K_;       // valid for tid < QT
    const float* bx = xyz + (size_t)b * N_ * 3;

    for (int ch = 0; ch < N_ / CHUNK; ++ch) {
        const int col = ch * CHUNK + (w << 4) + m0;
        const float* pp = bx + (size_t)col * 3;
        const float fx = pp[0], fy = pp[1], fz = pp[2];
        const float x2 = fx * fx + fy * fy + fz * fz;

        // B operand: lanes 0-15 : K=0,1 -> (px,py); lanes 16-31 : K=2,3 -> (pz,0)
        v2f Bv;
        Bv.x = (lane < 16) ? fx : fz;
        Bv.y = (lane < 16) ? fy : 0.0f;

        v8f acc = {0.f, 0.f, 0.f, 0.f, 0.f, 0.f, 0.f, 0.f};
        acc = __builtin_amdgcn_wmma_f32_16x16x4_f32(
            /*neg_a=*/false, Av, /*neg_b=*/false, Bv,
            /*c_mod=*/(short)0, acc, /*reuse_a=*/false, /*reuse_b=*/false);

        // D layout: VGPR j -> row (j + half8), column = lane&15
#pragma unroll
        for (int j = 0; j < 8; ++j) {
            const float sqr = q2r[j] + x2 - 2.0f * acc[j];
            s_mask[(j + half8) * CHUNK + (w << 4) + m0] = (sqr <= R2_) ? 1 : 0;
        }
        __syncthreads();

        // ordered first-K scan: thread m owns query m
        if (tid < QT && cnt < K_) {
            const unsigned char* mr = &s_mask[tid * CHUNK];
            for (int cc = 0; cc < CHUNK && cnt < K_; ++cc) {
                if (mr[cc]) { bq_idx[base + cnt] = ch * CHUNK + cc; ++cnt; }
            }
            s_cnt[tid] = cnt;
        }
        __syncthreads();
        if (tid == 0) {
            int done = 1;
            for (int m = 0; m < QT; ++m) done &= (s_cnt[m] >= K_);
            s_done = done;
        }
        __syncthreads();
        if (s_done) break;                    // uniform
    }

    // pad: entries beyond cnt repeat the first neighbor (reference semantics)
    if (tid < QT) {
        if (cnt == 0) {
            for (int k = 0; k < K_; ++k) bq_idx[base + k] = N_ - 1;
        } else {
            const int first = bq_idx[base];
            for (int k = cnt; k < K_; ++k) bq_idx[base + k] = first;
        }
    }
}

// ---------------------------------------------------------------------------
// Kernel 3: gather + affine + max over K, transposed output (B,C,S)
// ---------------------------------------------------------------------------
__global__ __launch_bounds__(C_)
void group_kernel(const float* __restrict__ points,
                  const int* __restrict__ fps_idx,
                  const int* __restrict__ bq_idx,
                  const float* __restrict__ alpha,
                  const float* __restrict__ beta,
                  float* __restrict__ new_points)
{
    const int blk = blockIdx.x;                // b*S + s
    const int b = blk / S_, s = blk % S_;
    const int c = threadIdx.x;

    const int a_idx     = fps_idx[blk];
    const float anchor  = points[((size_t)b * N_ + a_idx) * C_ + c];
    const float al      = alpha[c];
    const float be      = beta[c];
    const int* bq       = bq_idx + (size_t)blk * K_;

    float m = -INFINITY;
#pragma unroll
    for (int k = 0; k < K_; ++k) {
        const int idx = bq[k];
        const float v = points[((size_t)b * N_ + idx) * C_ + c];
        m = fmaxf(m, al * (v - anchor) + be);
    }
    new_points[((size_t)b * C_ + c) * S_ + s] = m;
}

// ---------------------------------------------------------------------------
extern "C" void kernel_launch(void* const* d_in, const int* in_sizes, int n_in,
                              void* d_out, int out_size, void* d_ws, size_t ws_size,
                              hipStream_t stream)
{
    const float* xyz    = (const float*)d_in[0];   // (B,N,3)
    const float* points = (const float*)d_in[1];   // (B,N,C)
    const float* alpha  = (const float*)d_in[2];   // (1,1,1,C)
    const float* beta   = (const float*)d_in[3];   // (1,1,1,C)

    float* out        = (float*)d_out;
    float* new_xyz    = out;                        // B*S*3 floats
    float* new_points = out + (size_t)B_ * S_ * 3;  // B*C*S floats

    int* fps_idx = (int*)d_ws;                      // B*S
    int* bq_idx  = fps_idx + (size_t)B_ * S_;       // B*S*K

    fps_kernel<<<B_, FPS_THREADS, 0, stream>>>(xyz, fps_idx, new_xyz);
    ballq_kernel<<<B_ * (S_ / QT), 256, 0, stream>>>(xyz, new_xyz, bq_idx);
    group_kernel<<<B_ * S_, C_, 0, stream>>>(points, fps_idx, bq_idx,
                                             alpha, beta, new_points);
}